// Correlation_80788334838106
// MI455X (gfx1250) — compile-verified
//
#include <hip/hip_runtime.h>

typedef __attribute__((ext_vector_type(2))) float v2f;
typedef __attribute__((ext_vector_type(8))) float v8f;

// D = A(16x4 f32) * B(4x16 f32) + C(16x16 f32), wave32, full fp32 precision.
static __device__ __forceinline__ v8f wmma_f32_16x16x4(v2f a, v2f b, v8f c) {
  // (neg_a, A, neg_b, B, c_mod, C, reuse_a, reuse_b) — probe-confirmed arity.
  return __builtin_amdgcn_wmma_f32_16x16x4_f32(
      false, a, false, b, (short)0, c, false, false);
}

#define MD     4                 // max displacement
#define DD     9                 // 2*MD+1
#define C_     64
#define H_     256
#define W_     256
#define TILE_W 64                // pixels per workgroup
#define QV     (TILE_W + 2*MD)   // 72 valid padded columns
#define Q_W    80                // + zero tail so B2 tile never reads OOB
#define CS     66                // padded channel stride: even (8B-aligned b64),
                                 // (2q+c)%64 bank pattern -> conflict-free fills

__global__ __launch_bounds__(128) void corr81_wmma_kernel(
    const float* __restrict__ x1, const float* __restrict__ x2,
    float* __restrict__ out) {
  __shared__ __align__(16) float ldsA[TILE_W][CS];    // x1 tile, [w][c]
  __shared__ __align__(16) float ldsB[Q_W][CS];       // x2 row,  [q][c]
  __shared__ __align__(16) float ldsD[4][2][16][17];  // [strip][tile][col][row]

  const int tid  = threadIdx.x;
  const int w0   = blockIdx.x * TILE_W;
  const int h    = blockIdx.y;
  const int b    = blockIdx.z;
  const int lane = tid & 31;
  const int warp = tid >> 5;   // 4 waves, one 16-pixel strip each
  const int jcol = lane & 15;  // fragment N (== M for A) position
  const int half = lane >> 4;  // selects K-pair (A/B) and M-half (C/D)
  const int wsub = warp * 16;

  // ---- stage x1[b,:,h,w0:w0+64] -> ldsA[w][c] (coalesced b128 global reads) ----
  {
    const long rowbase = (long)(b * C_) * (H_ * W_) + h * W_ + w0;
    for (int idx = tid; idx < C_ * (TILE_W / 4); idx += 128) {
      const int c  = idx >> 4;
      const int w4 = (idx & 15) << 2;
      const float4 v = *reinterpret_cast<const float4*>(
          &x1[rowbase + (long)c * (H_ * W_) + w4]);
      ldsA[w4 + 0][c] = v.x;
      ldsA[w4 + 1][c] = v.y;
      ldsA[w4 + 2][c] = v.z;
      ldsA[w4 + 3][c] = v.w;
    }
  }
  __syncthreads();

  // ---- preload all 16 A fragments once; reused for all 9 dx ----
  v2f aF[16];
#pragma unroll
  for (int k = 0; k < 16; ++k)
    aF[k] = *reinterpret_cast<const v2f*>(&ldsA[wsub + jcol][4 * k + 2 * half]);

  for (int dx = 0; dx < DD; ++dx) {
    const int h2 = h + dx - MD;
    const bool rowOk = (h2 >= 0) && (h2 < H_);

    // ---- stage zero-padded x2[b,:,h2,w0-4:w0+68] -> ldsB[q][c] ----
    for (int idx = tid; idx < C_ * Q_W; idx += 128) {
      const int c = idx / Q_W;
      const int q = idx - c * Q_W;  // consecutive tid -> consecutive q/p
      const int p = w0 - MD + q;
      float v = 0.0f;
      if (rowOk && q < QV && p >= 0 && p < W_)
        v = x2[(long)(b * C_ + c) * (H_ * W_) + h2 * W_ + p];
      ldsB[q][c] = v;
    }

    // hint the next dx row toward the caches (global_prefetch_b8)
    if ((dx + 1 < DD) && (h2 + 1 >= 0) && (h2 + 1 < H_)) {
      const int c = tid >> 1;
      int p = w0 - MD + (tid & 1) * 36;
      p = p < 0 ? 0 : (p > W_ - 1 ? W_ - 1 : p);
      __builtin_prefetch(&x2[(long)(b * C_ + c) * (H_ * W_) + (h2 + 1) * W_ + p],
                         0, 0);
    }
    __syncthreads();  // A: ldsB ready

    // ---- two 16x16 fp32 accumulators over K = 64 channels ----
    // acc1 columns: q = wsub + n ; acc2 columns: q = wsub + 16 + n
    v8f acc1 = {};
    v8f acc2 = {};
#pragma unroll
    for (int k = 0; k < 16; ++k) {
      const int c0 = 4 * k + 2 * half;
      const v2f b1 = *reinterpret_cast<const v2f*>(&ldsB[wsub + jcol][c0]);
      const v2f b2 = *reinterpret_cast<const v2f*>(&ldsB[wsub + 16 + jcol][c0]);
      acc1 = wmma_f32_16x16x4(aF[k], b1, acc1);
      acc2 = wmma_f32_16x16x4(aF[k], b2, acc2);
    }

    // ---- stage accumulators to LDS: lane holds col jcol, rows 8*half+v ----
#pragma unroll
    for (int v = 0; v < 8; ++v) {
      ldsD[warp][0][jcol][8 * half + v] = acc1[v];
      ldsD[warp][1][jcol][8 * half + v] = acc2[v];
    }
    __syncthreads();  // B: ldsD ready (also: all ldsB reads complete)

    // ---- coalesced band readback: out[b, dx*9+dy, h, w0 + i] ----
    // out(i,dy) = D[strip][ (ii+dy)>>4 ][ (ii+dy)&15 ][ ii ], ii = i&15
    for (int idx = tid; idx < DD * TILE_W; idx += 128) {
      const int dy    = idx >> 6;
      const int i     = idx & 63;
      const int strip = i >> 4;
      const int ii    = i & 15;
      const int j     = ii + dy;
      const float val = ldsD[strip][j >> 4][j & 15][ii];
      const long  o   = ((long)(b * (DD * DD) + dx * DD + dy) * H_ + h) * W_ +
                        (w0 + i);
      out[o] = val * 0.015625f;  // mean over 64 channels
    }
    // next iteration's barrier A orders ldsB refill and ldsD rewrite
  }
}

extern "C" void kernel_launch(void* const* d_in, const int* in_sizes, int n_in,
                              void* d_out, int out_size, void* d_ws,
                              size_t ws_size, hipStream_t stream) {
  (void)in_sizes; (void)n_in; (void)out_size; (void)d_ws; (void)ws_size;
  const float* x1 = (const float*)d_in[0];
  const float* x2 = (const float*)d_in[1];
  float* out = (float*)d_out;
  dim3 grid(W_ / TILE_W, H_, 8);  // (w-tiles, H, B)
  dim3 block(128);                // 4 wave32
  hipLaunchKernelGGL(corr81_wmma_kernel, grid, block, 0, stream, x1, x2, out);
}